// Attention_20890720928333
// MI455X (gfx1250) — compile-verified
//
#include <hip/hip_runtime.h>
#include <math.h>

#define B_ 64
#define N_ 197
#define D_ 768
#define H_ 12
#define DH_ 64
#define NP 208               // N padded to 13*16
#define NT 13                // number of 16-row/col tiles over NP
#define MASK_VAL -987654321.0f

typedef float v2f __attribute__((ext_vector_type(2)));
typedef float v8f __attribute__((ext_vector_type(8)));

__device__ __forceinline__ v8f wmma4(v2f a, v2f b, v8f c) {
  // D = A(16x4 f32) * B(4x16 f32) + C(16x16 f32)
  return __builtin_amdgcn_wmma_f32_16x16x4_f32(false, a, false, b, (short)0, c,
                                               false, false);
}

// ---------------------------------------------------------------------------
// Shared GEMM machinery: C[m,n] = sum_k A[m,k] * W[n,k]
// Block: 256 thr (8 waves), tile 128(M) x 128(N), K staged 16 at a time in
// LDS with register double-buffering of the global loads.
// ---------------------------------------------------------------------------
#define KT 16
#define LSTR 20   // LDS row stride (floats): 16B aligned rows, bank-spread

// Per-thread staging slots: e = tid + i*256 (512 float4 slots per 128x16 tile)
#define GEMM_PREFETCH(Aptr, Wptr, ktv)                                        \
  {                                                                           \
    int gm0 = m0 + row0, gm1 = m0 + row1;                                     \
    xv0 = make_float4(0.f, 0.f, 0.f, 0.f);                                    \
    xv1 = make_float4(0.f, 0.f, 0.f, 0.f);                                    \
    if (gm0 < M) xv0 = *(const float4*)(Aptr + (size_t)gm0 * D_ + (ktv) + c40);\
    if (gm1 < M) xv1 = *(const float4*)(Aptr + (size_t)gm1 * D_ + (ktv) + c41);\
    wv0 = *(const float4*)(Wptr + (size_t)(n0 + row0) * D_ + (ktv) + c40);    \
    wv1 = *(const float4*)(Wptr + (size_t)(n0 + row1) * D_ + (ktv) + c41);    \
  }

#define GEMM_MAINLOOP(Aptr, Wptr)                                             \
  const int M = B_ * N_;                                                      \
  const int m0 = blockIdx.y * 128;                                            \
  const int n0 = blockIdx.x * 128;                                            \
  const int tid = threadIdx.x;                                                \
  const int wave = tid >> 5, lane = tid & 31;                                 \
  const int ln16 = lane & 15, hi = lane >> 4;                                 \
  const int row0 = tid >> 2, c40 = (tid & 3) << 2;                            \
  const int row1 = (tid + 256) >> 2, c41 = c40;                               \
  float4 xv0, xv1, wv0, wv1;                                                  \
  v8f c[8];                                                                   \
  _Pragma("unroll") for (int j = 0; j < 8; ++j) c[j] = (v8f){};               \
  GEMM_PREFETCH(Aptr, Wptr, 0)                                                \
  for (int kt = 0; kt < D_; kt += KT) {                                       \
    __syncthreads();                                                          \
    *(float4*)(&Xs[row0 * LSTR + c40]) = xv0;                                 \
    *(float4*)(&Xs[row1 * LSTR + c41]) = xv1;                                 \
    *(float4*)(&Ws[row0 * LSTR + c40]) = wv0;                                 \
    *(float4*)(&Ws[row1 * LSTR + c41]) = wv1;                                 \
    __syncthreads();                                                          \
    if (kt + KT < D_) GEMM_PREFETCH(Aptr, Wptr, kt + KT)                      \
    _Pragma("unroll") for (int kk = 0; kk < 4; ++kk) {                        \
      int kb = kk * 4 + hi * 2;                                               \
      v2f a = *(const v2f*)(&Xs[(wave * 16 + ln16) * LSTR + kb]);             \
      v2f bfr[8];                                                             \
      _Pragma("unroll") for (int j = 0; j < 8; ++j)                           \
        bfr[j] = *(const v2f*)(&Ws[(j * 16 + ln16) * LSTR + kb]);             \
      _Pragma("unroll") for (int j = 0; j < 8; ++j)                           \
        c[j] = wmma4(a, bfr[j], c[j]);                                        \
    }                                                                         \
  }

// ---------------------------------------------------------------------------
// Kernel 1: QKV projection; scatter-stores into padded [B,H,NP,DH] buffers.
// ---------------------------------------------------------------------------
__global__ __launch_bounds__(256) void qkv_gemm(const float* __restrict__ X,
                                                const float* __restrict__ W,
                                                float* __restrict__ Q,
                                                float* __restrict__ K,
                                                float* __restrict__ V) {
  __shared__ float Xs[128 * LSTR];
  __shared__ float Ws[128 * LSTR];
  GEMM_MAINLOOP(X, W)

#pragma unroll
  for (int j = 0; j < 8; ++j) {
    int n = n0 + j * 16 + ln16;
    int which = n / D_;          // 0=q 1=k 2=v
    int rem = n - which * D_;
    int h = rem >> 6, d = rem & 63;
    float* outp = (which == 0) ? Q : ((which == 1) ? K : V);
#pragma unroll
    for (int vg = 0; vg < 8; ++vg) {
      int m = m0 + wave * 16 + vg + 8 * hi;   // C layout: M = vg + 8*(lane/16)
      if (m < M) {
        int b = m / N_, r = m - b * N_;
        outp[((size_t)(b * H_ + h) * NP + r) * DH_ + d] = c[j][vg];
      }
    }
  }
}

// ---------------------------------------------------------------------------
// Kernel 3: output projection. Y[m,n] = sum_k A[m,k]*Wout[n,k] + bias[n]
// ---------------------------------------------------------------------------
__global__ __launch_bounds__(256) void out_gemm(const float* __restrict__ A,
                                                const float* __restrict__ W,
                                                const float* __restrict__ bias,
                                                float* __restrict__ Y) {
  __shared__ float Xs[128 * LSTR];
  __shared__ float Ws[128 * LSTR];
  GEMM_MAINLOOP(A, W)

#pragma unroll
  for (int j = 0; j < 8; ++j) {
    int n = n0 + j * 16 + ln16;
    float bn = bias[n];
#pragma unroll
    for (int vg = 0; vg < 8; ++vg) {
      int m = m0 + wave * 16 + vg + 8 * hi;
      if (m < M) Y[(size_t)m * D_ + n] = c[j][vg] + bn;
    }
  }
}

// ---------------------------------------------------------------------------
// Kernel 2: per-(b,h) attention with LSA diagonal mask + softmax.
// K row-major (stride 66) and V *transposed* (stride 210) in LDS; per-wave P
// buffer (stride 210) converts softmax output from WMMA C-layout to A-layout.
// Q A-fragments preloaded into registers once per query tile.
// ---------------------------------------------------------------------------
#define KVSTR 66
#define PSTR 210
#define VTSTR 210

__global__ __launch_bounds__(256) void attn_kernel(const float* __restrict__ Q,
                                                   const float* __restrict__ K,
                                                   const float* __restrict__ V,
                                                   const float* __restrict__ scale,
                                                   float* __restrict__ Att) {
  extern __shared__ float smem[];
  float* Kl = smem;                       // NP * 66       (K row-major)
  float* Vt = Kl + NP * KVSTR;            // 64 * 210      (V transposed)
  float* Pl = Vt + DH_ * VTSTR;           // 8 waves * 16 * 210

  const int bh = blockIdx.x;
  const int b = bh / H_, h = bh - b * H_;
  const float sc = scale[h];
  const float* qp = Q + (size_t)bh * NP * DH_;
  const float* kp = K + (size_t)bh * NP * DH_;
  const float* vp = V + (size_t)bh * NP * DH_;
  const int tid = threadIdx.x;

  for (int idx = tid; idx < NP * DH_; idx += 256) {
    int r = idx >> 6, d = idx & 63;
    float kv = (r < N_) ? kp[idx] : 0.0f;
    float vv = (r < N_) ? vp[idx] : 0.0f;
    Kl[r * KVSTR + d] = kv;               // [row][d]
    Vt[d * VTSTR + r] = vv;               // [d][row]  (transposed)
  }
  __syncthreads();

  const int wave = tid >> 5, lane = tid & 31;
  const int ln16 = lane & 15, hi = lane >> 4;
  float* Pw = Pl + wave * (16 * PSTR);
  const float NEG_INF = -__builtin_inff();

  for (int tt = 0; tt < 2; ++tt) {
    const int qt = wave + tt * 8;          // query tile index (wave-uniform)
    if (qt >= NT) break;
    const float* qrow = qp + (size_t)(qt * 16 + ln16) * DH_;

    // ---- preload Q A-fragments for this tile (16 x v2f = 32 VGPRs) ----
    v2f aq[16];
#pragma unroll
    for (int kk = 0; kk < 16; ++kk)
      aq[kk] = *(const v2f*)(qrow + kk * 4 + hi * 2);

    // ---- S = Q K^T (13 tiles of 16x16, K=64): load burst then WMMA burst --
    v8f cs[NT];
#pragma unroll
    for (int j = 0; j < NT; ++j) {
      const float* krow = &Kl[(j * 16 + ln16) * KVSTR + hi * 2];
      v2f bf[16];
#pragma unroll
      for (int kk = 0; kk < 16; ++kk)
        bf[kk] = *(const v2f*)(krow + kk * 4);
      v8f acc = (v8f){};
#pragma unroll
      for (int kk = 0; kk < 16; ++kk) acc = wmma4(aq[kk], bf[kk], acc);
      cs[j] = acc;
    }

    // ---- scale, diagonal mask, pad mask, row max ----
    float mx[8];
#pragma unroll
    for (int vg = 0; vg < 8; ++vg) mx[vg] = NEG_INF;
#pragma unroll
    for (int j = 0; j < NT; ++j) {
      int col = j * 16 + ln16;
#pragma unroll
      for (int vg = 0; vg < 8; ++vg) {
        int row = qt * 16 + vg + 8 * hi;
        float s = cs[j][vg] * sc;
        if (col >= N_) s = NEG_INF;
        else if (col == row) s = MASK_VAL;
        cs[j][vg] = s;
        mx[vg] = fmaxf(mx[vg], s);
      }
    }
#pragma unroll
    for (int vg = 0; vg < 8; ++vg) {
#pragma unroll
      for (int off = 1; off < 16; off <<= 1)
        mx[vg] = fmaxf(mx[vg], __shfl_xor(mx[vg], off, 32));
    }

    // ---- exp + row sum ----
    float sm[8];
#pragma unroll
    for (int vg = 0; vg < 8; ++vg) sm[vg] = 0.0f;
#pragma unroll
    for (int j = 0; j < NT; ++j) {
#pragma unroll
      for (int vg = 0; vg < 8; ++vg) {
        float p = __expf(cs[j][vg] - mx[vg]);
        cs[j][vg] = p;
        sm[vg] += p;
      }
    }
#pragma unroll
    for (int vg = 0; vg < 8; ++vg) {
#pragma unroll
      for (int off = 1; off < 16; off <<= 1)
        sm[vg] += __shfl_xor(sm[vg], off, 32);
    }
    float rs[8];
#pragma unroll
    for (int vg = 0; vg < 8; ++vg) rs[vg] = 1.0f / sm[vg];

    // ---- P -> per-wave LDS buffer (C-layout to row-major) ----
#pragma unroll
    for (int j = 0; j < NT; ++j) {
#pragma unroll
      for (int vg = 0; vg < 8; ++vg)
        Pw[(vg + 8 * hi) * PSTR + j * 16 + ln16] = cs[j][vg];
    }
    // same-wave DS ops are in-order; buffer is wave-private -> no barrier

    // ---- O = P V (4 tiles along DH, K = 208) ----
#pragma unroll
    for (int dt = 0; dt < 4; ++dt) {
      const float* prow = &Pw[ln16 * PSTR + hi * 2];
      const float* vrow = &Vt[(dt * 16 + ln16) * VTSTR + hi * 2];
      v8f co = (v8f){};
#pragma unroll 4
      for (int jk = 0; jk < NP / 4; ++jk) {
        v2f a = *(const v2f*)(prow + jk * 4);
        v2f bf = *(const v2f*)(vrow + jk * 4);   // single b64: V is transposed
        co = wmma4(a, bf, co);
      }
#pragma unroll
      for (int vg = 0; vg < 8; ++vg) {
        int row = qt * 16 + vg + 8 * hi;
        if (row < N_)
          Att[((size_t)(b * N_ + row)) * D_ + h * DH_ + dt * 16 + ln16] =
              co[vg] * rs[vg];
      }
    }
  }
}

// ---------------------------------------------------------------------------
extern "C" void kernel_launch(void* const* d_in, const int* in_sizes, int n_in,
                              void* d_out, int out_size, void* d_ws,
                              size_t ws_size, hipStream_t stream) {
  (void)in_sizes; (void)n_in; (void)out_size; (void)ws_size;
  const float* x     = (const float*)d_in[0];  // [B,N,D]
  const float* Wqkv  = (const float*)d_in[1];  // [2304,768]
  const float* scale = (const float*)d_in[2];  // [H]
  const float* Wout  = (const float*)d_in[3];  // [768,768]
  const float* bout  = (const float*)d_in[4];  // [768]
  float* y = (float*)d_out;                    // [B,N,D]

  float* ws = (float*)d_ws;
  const size_t QSZ = (size_t)B_ * H_ * NP * DH_;  // padded per-tensor size
  float* q   = ws;
  float* k   = ws + QSZ;
  float* v   = ws + 2 * QSZ;
  float* att = ws + 3 * QSZ;                   // [B*N, 768]

  dim3 g1(3 * D_ / 128, (B_ * N_ + 127) / 128);
  qkv_gemm<<<g1, 256, 0, stream>>>(x, Wqkv, q, k, v);

  size_t smem =
      (size_t)(NP * KVSTR + DH_ * VTSTR + 8 * 16 * PSTR) * sizeof(float);
  attn_kernel<<<dim3(B_ * H_), 256, smem, stream>>>(q, k, v, scale, att);

  dim3 g3(D_ / 128, (B_ * N_ + 127) / 128);
  out_gemm<<<g3, 256, 0, stream>>>(att, Wout, bout, y);
}